// TestModule_43361989820886
// MI455X (gfx1250) — compile-verified
//
#include <hip/hip_runtime.h>
#include <hip/hip_bf16.h>

typedef __attribute__((ext_vector_type(2))) float v2f;
typedef __attribute__((ext_vector_type(8))) float v8f;

#define N_NODES   50000
#define N_EDGES   1600000
#define IN_CH     64
#define HID       16
#define N_CLASSES 10
#define NTILES    (N_NODES / 16)   // 3125 exact, no tail

// ---------------------------------------------------------------------------
// Zero-fill scratch (aggregation accumulators) — must run every call since the
// harness does not re-initialize d_ws between replays.
// ---------------------------------------------------------------------------
__global__ void zero_kernel(float* __restrict__ p, long long n) {
    long long i = (long long)blockIdx.x * blockDim.x + threadIdx.x;
    if (i < n) p[i] = 0.0f;
}

// ---------------------------------------------------------------------------
// Layer 1 projections via V_WMMA_F32_16X16X4_F32.
// One wave (32 lanes) per 16-node tile. K = 64 -> 16 WMMA steps per output,
// two outputs (W_rel, W_root) -> 32 WMMAs per wave.
// A (16x4 f32): lane m = lane&15, half = lane>>4 selects K pair {0,1}/{2,3}.
// B (4x16 f32): lane n = lane&15, same half/K-pair mapping.
// C/D (16x16 f32): VGPR v holds row (v + 8*half), col = lane&15.
// ---------------------------------------------------------------------------
__global__ __launch_bounds__(32) void gemm1_wmma(
    const float* __restrict__ x,
    const float* __restrict__ w_rel,   // [64,16] row-major
    const float* __restrict__ w_root,  // [64,16] row-major
    float* __restrict__ t1,            // [N,16] x @ w_rel
    float* __restrict__ r1)            // [N,16] x @ w_root
{
    const int lane = threadIdx.x;
    const int tile = blockIdx.x;
    const int n    = lane & 15;
    const int half = lane >> 4;
    const float* xrow = x + (size_t)(tile * 16 + n) * IN_CH;  // A row M = lane&15

    v8f acc_rel = {};
    v8f acc_rt  = {};
#pragma unroll
    for (int k0 = 0; k0 < IN_CH; k0 += 4) {
        const int ka = k0 + half * 2;
        v2f a;    a.x    = xrow[ka];              a.y    = xrow[ka + 1];
        v2f brel; brel.x = w_rel[ka * HID + n];   brel.y = w_rel[(ka + 1) * HID + n];
        v2f brt;  brt.x  = w_root[ka * HID + n];  brt.y  = w_root[(ka + 1) * HID + n];
        acc_rel = __builtin_amdgcn_wmma_f32_16x16x4_f32(
            false, a, false, brel, (short)0, acc_rel, false, false);
        acc_rt = __builtin_amdgcn_wmma_f32_16x16x4_f32(
            false, a, false, brt, (short)0, acc_rt, false, false);
    }
#pragma unroll
    for (int v = 0; v < 8; ++v) {
        const int mr = v + half * 8;
        const size_t o = (size_t)(tile * 16 + mr) * HID + n;
        t1[o] = acc_rel[v];
        r1[o] = acc_rt[v];
    }
}

// ---------------------------------------------------------------------------
// Layer 2 projections: [N,16] @ [16,10]. N padded to 16 with zero B columns
// (accumulator cols 10..15 stay zero and are never consumed).
// ---------------------------------------------------------------------------
__global__ __launch_bounds__(32) void gemm2_wmma(
    const float* __restrict__ h,
    const float* __restrict__ w_rel,   // [16,10] row-major
    const float* __restrict__ w_root,  // [16,10] row-major
    float* __restrict__ t2,            // [N,16] (cols>=10 zero)
    float* __restrict__ r2)            // [N,16] (cols>=10 zero)
{
    const int lane = threadIdx.x;
    const int tile = blockIdx.x;
    const int n    = lane & 15;
    const int half = lane >> 4;
    const float* hrow = h + (size_t)(tile * 16 + n) * HID;

    v8f acc_rel = {};
    v8f acc_rt  = {};
#pragma unroll
    for (int k0 = 0; k0 < HID; k0 += 4) {
        const int ka = k0 + half * 2;
        v2f a; a.x = hrow[ka]; a.y = hrow[ka + 1];
        v2f brel = {};
        v2f brt  = {};
        if (n < N_CLASSES) {  // divergence only on loads; EXEC restored pre-WMMA
            brel.x = w_rel[ka * N_CLASSES + n];
            brel.y = w_rel[(ka + 1) * N_CLASSES + n];
            brt.x  = w_root[ka * N_CLASSES + n];
            brt.y  = w_root[(ka + 1) * N_CLASSES + n];
        }
        acc_rel = __builtin_amdgcn_wmma_f32_16x16x4_f32(
            false, a, false, brel, (short)0, acc_rel, false, false);
        acc_rt = __builtin_amdgcn_wmma_f32_16x16x4_f32(
            false, a, false, brt, (short)0, acc_rt, false, false);
    }
#pragma unroll
    for (int v = 0; v < 8; ++v) {
        const int mr = v + half * 8;
        const size_t o = (size_t)(tile * 16 + mr) * HID + n;
        t2[o] = acc_rel[v];
        r2[o] = acc_rt[v];
    }
}

// ---------------------------------------------------------------------------
// Edge scatter-add, 16 channels per edge. One thread per (edge, 4-channel
// group): float4 gather (global_load_b128) + 4x global_atomic_add_f32.
// ---------------------------------------------------------------------------
__global__ void edge_agg16(const long long* __restrict__ src,
                           const long long* __restrict__ dst,
                           const float* __restrict__ feat,  // [N,16]
                           float* __restrict__ agg)         // [N,16]
{
    const long long idx = (long long)blockIdx.x * blockDim.x + threadIdx.x;
    if (idx >= (long long)N_EDGES * 4) return;
    const int e = (int)(idx >> 2);
    const int g = (int)(idx & 3);
    const int s = (int)src[e];
    const int d = (int)dst[e];
    const float4 v = *(const float4*)(feat + (size_t)s * 16 + g * 4);
    float* ap = agg + (size_t)d * 16 + g * 4;
    atomicAdd(ap + 0, v.x);
    atomicAdd(ap + 1, v.y);
    atomicAdd(ap + 2, v.z);
    atomicAdd(ap + 3, v.w);
}

// Edge scatter-add, 10 active channels of stride-16 rows.
__global__ void edge_agg10(const long long* __restrict__ src,
                           const long long* __restrict__ dst,
                           const float* __restrict__ feat,  // [N,16], cols<10 valid
                           float* __restrict__ agg)         // [N,16]
{
    const long long idx = (long long)blockIdx.x * blockDim.x + threadIdx.x;
    if (idx >= (long long)N_EDGES * 16) return;
    const int e = (int)(idx >> 4);
    const int c = (int)(idx & 15);
    if (c >= N_CLASSES) return;
    const int s = (int)src[e];
    const int d = (int)dst[e];
    atomicAdd(agg + (size_t)d * 16 + c, feat[(size_t)s * 16 + c]);
}

// h = relu(agg1 + r1 + b_rel1)
__global__ void combine_relu(const float* __restrict__ agg1,
                             const float* __restrict__ r1,
                             const float* __restrict__ b,
                             float* __restrict__ h)
{
    const int idx = blockIdx.x * blockDim.x + threadIdx.x;
    if (idx >= N_NODES * HID) return;
    const int c = idx & 15;
    const float v = agg1[idx] + r1[idx] + b[c];
    h[idx] = fmaxf(v, 0.0f);
}

// out = log_softmax(agg2 + r2 + b_rel2), 10 classes per thread.
__global__ void final_logsoftmax(const float* __restrict__ agg2,
                                 const float* __restrict__ r2,
                                 const float* __restrict__ b,
                                 float* __restrict__ out)
{
    const int i = blockIdx.x * blockDim.x + threadIdx.x;
    if (i >= N_NODES) return;
    float vals[N_CLASSES];
    float mx = -3.402823466e+38f;
#pragma unroll
    for (int c = 0; c < N_CLASSES; ++c) {
        const float v = agg2[(size_t)i * 16 + c] + r2[(size_t)i * 16 + c] + b[c];
        vals[c] = v;
        mx = fmaxf(mx, v);
    }
    float s = 0.0f;
#pragma unroll
    for (int c = 0; c < N_CLASSES; ++c) s += expf(vals[c] - mx);
    const float lse = mx + logf(s);
#pragma unroll
    for (int c = 0; c < N_CLASSES; ++c)
        out[(size_t)i * N_CLASSES + c] = vals[c] - lse;
}

// ---------------------------------------------------------------------------
extern "C" void kernel_launch(void* const* d_in, const int* in_sizes, int n_in,
                              void* d_out, int out_size, void* d_ws, size_t ws_size,
                              hipStream_t stream) {
    const float*     x       = (const float*)d_in[0];
    const long long* ei      = (const long long*)d_in[1];   // int64 [2, E]
    const long long* src     = ei;
    const long long* dst     = ei + N_EDGES;
    const float*     w_rel1  = (const float*)d_in[2];
    const float*     b_rel1  = (const float*)d_in[3];
    const float*     w_root1 = (const float*)d_in[4];
    const float*     w_rel2  = (const float*)d_in[5];
    const float*     b_rel2  = (const float*)d_in[6];
    const float*     w_root2 = (const float*)d_in[7];
    float*           out     = (float*)d_out;

    // Workspace: 4 buffers of [N_NODES,16] f32 (3.2 MB each), reused:
    //   bufA: t1 -> h      bufB: r1 -> t2     bufC: agg1 -> r2     bufD: agg2
    const size_t NB = (size_t)N_NODES * 16;
    float* bufA = (float*)d_ws;
    float* bufB = bufA + NB;
    float* bufC = bufB + NB;
    float* bufD = bufC + NB;

    // Zero both aggregation accumulators (bufC, bufD are contiguous).
    {
        const long long n = 2 * (long long)NB;
        zero_kernel<<<(int)((n + 255) / 256), 256, 0, stream>>>(bufC, n);
    }

    // Layer 1: project (WMMA), aggregate projected features, combine+ReLU.
    gemm1_wmma<<<NTILES, 32, 0, stream>>>(x, w_rel1, w_root1, bufA, bufB);
    {
        const long long n = (long long)N_EDGES * 4;
        edge_agg16<<<(int)((n + 255) / 256), 256, 0, stream>>>(src, dst, bufA, bufC);
    }
    combine_relu<<<(N_NODES * HID + 255) / 256, 256, 0, stream>>>(bufC, bufB, b_rel1, bufA);

    // Layer 2: project (WMMA), aggregate, combine + log-softmax.
    gemm2_wmma<<<NTILES, 32, 0, stream>>>(bufA, w_rel2, w_root2, bufB, bufC);
    {
        const long long n = (long long)N_EDGES * 16;
        edge_agg10<<<(int)((n + 255) / 256), 256, 0, stream>>>(src, dst, bufB, bufD);
    }
    final_logsoftmax<<<(N_NODES + 255) / 256, 256, 0, stream>>>(bufD, bufC, b_rel2, out);
}